// OHNMLoss_42417097016427
// MI455X (gfx1250) — compile-verified
//
#include <hip/hip_runtime.h>
#include <math.h>

// ---------------------------------------------------------------------------
// OHEM BCE loss for MI455X (gfx1250, wave32).
//
// Strategy: 3-level radix-select (11+11+10 bits) over the monotonic
// float->uint mapping, with per-bin {count, softplus-sum} histograms built in
// LDS (ds_add_u32 / ds_add_f32) and flushed with global atomics. Positive-BCE
// sum is fused into pass 1. Each full sweep reads 67MB (x+y); the 192MB L2
// keeps passes 2/3 cache-resident. Pass 1 stages x through the CDNA5 async
// path (global_load_async_to_lds_b128 + s_wait_asynccnt); passes 2/3 issue
// global_prefetch_b8.
// ---------------------------------------------------------------------------

#define NELEM   (8u * 1024u * 1024u)   // B*H*W = 8388608
#define L1BINS  2048
#define L2BINS  2048
#define L3BINS  1024

// workspace layout, 4-byte words
#define WS_POSSUM 0
#define WS_NEGACC 1
#define WS_PREFIX 2
#define WS_KREM   3
#define WS_H1C    8
#define WS_H1S    (WS_H1C + L1BINS)
#define WS_H2C    (WS_H1S + L1BINS)
#define WS_H2S    (WS_H2C + L2BINS)
#define WS_H3C    (WS_H2S + L2BINS)
#define WS_H3S    (WS_H3C + L3BINS)
#define WS_WORDS  (WS_H3S + L3BINS)

#if defined(__HIP_DEVICE_COMPILE__) && __has_builtin(__builtin_amdgcn_global_load_async_to_lds_b128)
#define ASYNC_OK 1
typedef int v4i_t __attribute__((vector_size(16)));
typedef __attribute__((address_space(1))) v4i_t g_v4i;   // global (AS1)
typedef __attribute__((address_space(3))) v4i_t l_v4i;   // LDS (AS3)
#else
#define ASYNC_OK 0
#endif

__device__ __forceinline__ void wait_async0() {
#if defined(__HIP_DEVICE_COMPILE__)
#if __has_builtin(__builtin_amdgcn_s_wait_asynccnt)
  __builtin_amdgcn_s_wait_asynccnt(0);
#else
  asm volatile("s_wait_asynccnt 0" ::: "memory");
#endif
#endif
}

// order-preserving float -> uint map (larger float => larger uint)
__device__ __forceinline__ unsigned fmap(float f) {
  unsigned b = __float_as_uint(f);
  return (b & 0x80000000u) ? ~b : (b | 0x80000000u);
}
__device__ __forceinline__ float funmap(unsigned u) {
  unsigned b = (u & 0x80000000u) ? (u ^ 0x80000000u) : ~u;
  return __uint_as_float(b);
}
// BCE(x, 0) = softplus(x)
__device__ __forceinline__ float softplus_f(float x) {
  return fmaxf(x, 0.0f) + log1pf(expf(-fabsf(x)));
}
// BCE(x, 1)
__device__ __forceinline__ float bce_pos(float x) {
  return fmaxf(x, 0.0f) - x + log1pf(expf(-fabsf(x)));
}

__global__ __launch_bounds__(256) void k_zero(unsigned* ws, int nwords) {
  int i = blockIdx.x * 256 + threadIdx.x;
  if (i < nwords) ws[i] = 0u;
}

// ---------------------------------------------------------------------------
// Pass 1: level-1 histogram (top 11 bits of mapped value) + positive BCE sum
// ---------------------------------------------------------------------------
__device__ __forceinline__ void proc_l1(float xc, float yc, float& pacc,
                                        unsigned* sc, float* ss) {
  if (yc > 0.0f) {
    pacc += bce_pos(xc);
  } else {
    unsigned u = fmap(xc);
    unsigned b = u >> 21;
    atomicAdd(&sc[b], 1u);                 // ds_add_u32
    atomicAdd(&ss[b], softplus_f(xc));     // ds_add_f32
  }
}

__global__ __launch_bounds__(256) void k_pass1(const float* __restrict__ x,
                                               const float* __restrict__ y,
                                               unsigned* hc, float* hs,
                                               float* possum) {
  __shared__ unsigned sc[L1BINS];
  __shared__ float    ss[L1BINS];
#if ASYNC_OK
  __shared__ float    xtile[256 * 4];      // one float4 staging slot / thread
#endif
  __shared__ float    red[256];

  for (int i = threadIdx.x; i < L1BINS; i += 256) { sc[i] = 0u; ss[i] = 0.0f; }
  __syncthreads();

  const float4* x4 = (const float4*)x;
  const float4* y4 = (const float4*)y;
  const unsigned n4 = NELEM / 4;
  const unsigned stride = gridDim.x * 256u;
  float pacc = 0.0f;

  for (unsigned i = blockIdx.x * 256u + threadIdx.x; i < n4; i += stride) {
    float4 yv = y4[i];
    float4 xv;
#if ASYNC_OK
    // CDNA5 async path: DMA 16B/lane global->LDS, wait asynccnt, read back.
    float* slot = &xtile[threadIdx.x * 4];
    __builtin_amdgcn_global_load_async_to_lds_b128(
        (g_v4i*)(x4 + i), (l_v4i*)slot, 0, 0);
    wait_async0();
    volatile float* vslot = (volatile float*)slot;
    xv.x = vslot[0]; xv.y = vslot[1]; xv.z = vslot[2]; xv.w = vslot[3];
#else
    xv = x4[i];
#endif
    proc_l1(xv.x, yv.x, pacc, sc, ss);
    proc_l1(xv.y, yv.y, pacc, sc, ss);
    proc_l1(xv.z, yv.z, pacc, sc, ss);
    proc_l1(xv.w, yv.w, pacc, sc, ss);
  }
  __syncthreads();

  for (int i = threadIdx.x; i < L1BINS; i += 256) {
    unsigned c = sc[i];
    if (c) atomicAdd(&hc[i], c);
    float s = ss[i];
    if (s != 0.0f) atomicAdd(&hs[i], s);
  }

  red[threadIdx.x] = pacc;
  __syncthreads();
  for (int s = 128; s > 0; s >>= 1) {
    if ((int)threadIdx.x < s) red[threadIdx.x] += red[threadIdx.x + s];
    __syncthreads();
  }
  if (threadIdx.x == 0) atomicAdd(possum, red[0]);
}

// ---------------------------------------------------------------------------
// Passes 2/3: refine within the selected prefix. cmpShift/binShift/binMask
// select which bits are compared vs histogrammed.
// ---------------------------------------------------------------------------
__global__ __launch_bounds__(256) void k_refine(const float* __restrict__ x,
                                                const float* __restrict__ y,
                                                const unsigned* prefix_p,
                                                unsigned* hc, float* hs,
                                                int cmpShift, int binShift,
                                                unsigned binMask, int nbins) {
  __shared__ unsigned sc[L2BINS];
  __shared__ float    ss[L2BINS];
  for (int i = threadIdx.x; i < nbins; i += 256) { sc[i] = 0u; ss[i] = 0.0f; }
  __syncthreads();

  const unsigned P = *prefix_p;            // uniform -> scalarized
  const float4* x4 = (const float4*)x;
  const float4* y4 = (const float4*)y;
  const unsigned n4 = NELEM / 4;
  const unsigned stride = gridDim.x * 256u;

  for (unsigned i = blockIdx.x * 256u + threadIdx.x; i < n4; i += stride) {
    __builtin_prefetch(&x4[i + stride], 0, 0);   // global_prefetch_b8
    __builtin_prefetch(&y4[i + stride], 0, 0);
    float4 xv = x4[i];
    float4 yv = y4[i];
    float xs[4] = {xv.x, xv.y, xv.z, xv.w};
    float ys[4] = {yv.x, yv.y, yv.z, yv.w};
#pragma unroll
    for (int c = 0; c < 4; ++c) {
      if (ys[c] == 0.0f) {
        unsigned u = fmap(xs[c]);
        if ((u >> cmpShift) == P) {
          unsigned b = (u >> binShift) & binMask;
          atomicAdd(&sc[b], 1u);
          atomicAdd(&ss[b], softplus_f(xs[c]));
        }
      }
    }
  }
  __syncthreads();
  for (int i = threadIdx.x; i < nbins; i += 256) {
    unsigned c = sc[i];
    if (c) atomicAdd(&hc[i], c);
    float s = ss[i];
    if (s != 0.0f) atomicAdd(&hs[i], s);
  }
}

// ---------------------------------------------------------------------------
// Single-block scan: find threshold bin (descending), accumulate above-sums.
// level 0: init k from pos*3. level 2: resolve ties + write final scalar.
// ---------------------------------------------------------------------------
__global__ __launch_bounds__(1024) void k_scan(const unsigned* hc,
                                               const float* hs, int nbins,
                                               unsigned* prefix, unsigned* krem,
                                               float* negacc,
                                               const int* pos_ptr, int level,
                                               const float* possum,
                                               float* out) {
  __shared__ unsigned sc[L1BINS];
  __shared__ float    ss[L1BINS];
  for (int i = threadIdx.x; i < nbins; i += 1024) { sc[i] = hc[i]; ss[i] = hs[i]; }
  __syncthreads();
  if (threadIdx.x != 0) return;

  const long long pos = (long long)(*pos_ptr);
  const long long kin = (level == 0) ? pos * 3 : (long long)(*krem);
  const unsigned  pin = (level == 0) ? 0u : *prefix;

  long long acc = 0;
  double sacc = 0.0;
  int b = 0;
  for (int i = nbins - 1; i >= 0; --i) {
    long long c = (long long)sc[i];
    if (acc + c >= kin) { b = i; break; }
    acc += c;
    sacc += (double)ss[i];
  }

  float negNew = *negacc + (float)sacc;
  *negacc = negNew;
  *krem = (unsigned)(kin - acc);

  if (level == 0) {
    *prefix = (unsigned)b;                       // 11-bit prefix
  } else if (level == 1) {
    *prefix = (pin << 11) | (unsigned)b;         // 22-bit prefix
  } else {
    unsigned u_exact = (pin << 10) | (unsigned)b;  // exact 32-bit threshold
    float t = funmap(u_exact);
    float tie = (float)(kin - acc) * softplus_f(t);
    float total = *possum + negNew + tie;
    out[0] = total / (float)(pos + pos * 3);
  }
}

// ---------------------------------------------------------------------------
extern "C" void kernel_launch(void* const* d_in, const int* in_sizes, int n_in,
                              void* d_out, int out_size, void* d_ws,
                              size_t ws_size, hipStream_t stream) {
  const float* x = (const float*)d_in[0];
  const float* y = (const float*)d_in[1];
  const int* posp = (const int*)d_in[2];
  float* out = (float*)d_out;
  unsigned* ws = (unsigned*)d_ws;
  float* wsf = (float*)d_ws;

  (void)in_sizes; (void)n_in; (void)out_size; (void)ws_size;

  k_zero<<<(WS_WORDS + 255) / 256, 256, 0, stream>>>(ws, WS_WORDS);

  // level 1: top 11 bits
  k_pass1<<<2048, 256, 0, stream>>>(x, y, ws + WS_H1C, wsf + WS_H1S,
                                    wsf + WS_POSSUM);
  k_scan<<<1, 1024, 0, stream>>>(ws + WS_H1C, wsf + WS_H1S, L1BINS,
                                 ws + WS_PREFIX, ws + WS_KREM, wsf + WS_NEGACC,
                                 posp, 0, wsf + WS_POSSUM, out);

  // level 2: next 11 bits within prefix (L2-resident sweep)
  k_refine<<<2048, 256, 0, stream>>>(x, y, ws + WS_PREFIX, ws + WS_H2C,
                                     wsf + WS_H2S, 21, 10, 0x7FFu, L2BINS);
  k_scan<<<1, 1024, 0, stream>>>(ws + WS_H2C, wsf + WS_H2S, L2BINS,
                                 ws + WS_PREFIX, ws + WS_KREM, wsf + WS_NEGACC,
                                 posp, 1, wsf + WS_POSSUM, out);

  // level 3: last 10 bits -> exact threshold, ties, final scalar
  k_refine<<<2048, 256, 0, stream>>>(x, y, ws + WS_PREFIX, ws + WS_H3C,
                                     wsf + WS_H3S, 10, 0, 0x3FFu, L3BINS);
  k_scan<<<1, 1024, 0, stream>>>(ws + WS_H3C, wsf + WS_H3S, L3BINS,
                                 ws + WS_PREFIX, ws + WS_KREM, wsf + WS_NEGACC,
                                 posp, 2, wsf + WS_POSSUM, out);
}